// GIN_MAS_2757369004100
// MI455X (gfx1250) — compile-verified
//
#include <hip/hip_runtime.h>
#include <hip/hip_bf16.h>

#define N_NODES 50000
#define N_EDGES 800000
#define HD      128
#define NLAYERS 3
#define NGRAPH  64
#define NODES_PER_BLOCK 400

typedef float v2f __attribute__((ext_vector_type(2)));
typedef float v8f __attribute__((ext_vector_type(8)));

// One wave (32 lanes) per edge: lane loads float4 of h[src] (512B/row total),
// atomically adds into agg[dst]. agg (25.6MB) lives in L2 -> L2-side atomics.
__global__ void __launch_bounds__(256) gin_edge_scatter(
    const float* __restrict__ h, const int* __restrict__ src,
    const int* __restrict__ dst, float* __restrict__ agg)
{
    int gtid = blockIdx.x * blockDim.x + threadIdx.x;
    int edge = gtid >> 5;
    int lane = gtid & 31;
    if (edge >= N_EDGES) return;
    int s = src[edge];
    int d = dst[edge];
    const float4 v = ((const float4*)(h + (size_t)s * HD))[lane];
    float* a = agg + (size_t)d * HD + (size_t)lane * 4;
    atomicAdd(a + 0, v.x);
    atomicAdd(a + 1, v.y);
    atomicAdd(a + 2, v.z);
    atomicAdd(a + 3, v.w);
}

// Fused GIN half-MLP: Y = act((X [+ Xadd]) @ W + bias)
// Block = 256 threads = 8 waves. Block owns one 16-row tile; wave w owns
// column tile w (16 cols). K=128 swept by 32 x v_wmma_f32_16x16x4_f32.
// W (128x128 = 64KB) + A-tile (16x128 = 8KB) staged in LDS (72KB/WGP, max 320KB).
__global__ void __launch_bounds__(256) gin_mlp_wmma(
    const float* __restrict__ X, const float* __restrict__ Xadd,
    const float* __restrict__ W, const float* __restrict__ bias,
    float* __restrict__ Y, int do_relu)
{
    __shared__ float Wlds[HD * HD];   // 64 KB
    __shared__ float Alds[16 * HD];   //  8 KB
    const int tid = threadIdx.x;
    const int rowtile = blockIdx.x;   // 0..3124 (50000/16 exact)

    // Stage weights (row-major [k][n]); W is L2-resident (64KB << 192MB L2).
    {
        const float4* Wg = (const float4*)W;
        float4* Wl = (float4*)Wlds;
        for (int i = tid; i < (HD * HD) / 4; i += 256) Wl[i] = Wg[i];
    }
    // Stage A tile with fused z = h + agg when Xadd != nullptr.
    {
        const float4* Xg = (const float4*)X + (size_t)rowtile * (16 * HD / 4);
        float4* Al = (float4*)Alds;
        if (Xadd) {
            const float4* Ag = (const float4*)Xadd + (size_t)rowtile * (16 * HD / 4);
            for (int i = tid; i < (16 * HD) / 4; i += 256) {
                float4 a = Xg[i], b = Ag[i];
                a.x += b.x; a.y += b.y; a.z += b.z; a.w += b.w;
                Al[i] = a;
            }
        } else {
            for (int i = tid; i < (16 * HD) / 4; i += 256) Al[i] = Xg[i];
        }
    }
    __syncthreads();

    const int wave = tid >> 5;        // column tile 0..7
    const int lane = tid & 31;
    const int half = lane >> 4;       // 0: K={k,k+1}, 1: K={k+2,k+3}
    const int l15  = lane & 15;
    const int col  = wave * 16 + l15; // output column (N index)
    const int koff = half * 2;

    // Bias folded into accumulator: D layout has N = lane&15 in every VGPR.
    v8f acc;
    const float bv = bias[col];
#pragma unroll
    for (int i = 0; i < 8; ++i) acc[i] = bv;

#pragma unroll
    for (int k = 0; k < HD; k += 4) {
        v2f a, b;
        // A 16x4 frag: lane half selects K pair (ISA 32-bit A layout).
        a.x = Alds[l15 * HD + k + koff];
        a.y = Alds[l15 * HD + k + koff + 1];
        // B 4x16 frag: one K-row striped across lanes per VGPR.
        b.x = Wlds[(k + koff) * HD + col];
        b.y = Wlds[(k + koff + 1) * HD + col];
        acc = __builtin_amdgcn_wmma_f32_16x16x4_f32(
            /*neg_a=*/false, a, /*neg_b=*/false, b,
            /*c_mod=*/(short)0, acc, /*reuse_a=*/false, /*reuse_b=*/false);
    }

    if (do_relu) {
#pragma unroll
        for (int i = 0; i < 8; ++i) acc[i] = fmaxf(acc[i], 0.0f);
    }

    // D layout: VGPR r -> M = r + 8*half; lanes write 64B runs per row.
    const int mbase = rowtile * 16 + half * 8;
#pragma unroll
    for (int r = 0; r < 8; ++r)
        Y[(size_t)(mbase + r) * HD + col] = acc[r];
}

// Per-graph sum readout. graph_ids sorted -> register accumulation with one
// atomic flush per graph transition per block. 128 threads = 1 feature each.
__global__ void __launch_bounds__(128) gin_readout(
    const float* __restrict__ h, const int* __restrict__ gids,
    float* __restrict__ ro)
{
    const int f = threadIdx.x;
    const int start = blockIdx.x * NODES_PER_BLOCK;
    int end = start + NODES_PER_BLOCK;
    if (end > N_NODES) end = N_NODES;
    float acc = 0.0f;
    int cur = gids[start];
    for (int i = start; i < end; ++i) {
        int g = gids[i];
        if (g != cur) {
            atomicAdd(&ro[(size_t)cur * HD + f], acc);
            acc = 0.0f;
            cur = g;
        }
        acc += h[(size_t)i * HD + f];
    }
    atomicAdd(&ro[(size_t)cur * HD + f], acc);
}

// out[g] = b_out + sum_{l,f} ro[l][g][f] * w_out[l*H + f]
__global__ void __launch_bounds__(64) gin_final(
    const float* __restrict__ ro, const float* __restrict__ w_out,
    const float* __restrict__ b_out, float* __restrict__ out)
{
    const int g = threadIdx.x;
    float acc = b_out[0];
    for (int l = 0; l < NLAYERS; ++l)
        for (int f = 0; f < HD; ++f)
            acc += ro[((size_t)l * NGRAPH + g) * HD + f] * w_out[l * HD + f];
    out[g] = acc;
}

extern "C" void kernel_launch(void* const* d_in, const int* in_sizes, int n_in,
                              void* d_out, int out_size, void* d_ws, size_t ws_size,
                              hipStream_t stream)
{
    const float* h0   = (const float*)d_in[0];
    const int*   src  = (const int*)d_in[1];
    const int*   dst  = (const int*)d_in[2];
    const int*   gids = (const int*)d_in[3];
    const float* w1[NLAYERS] = {(const float*)d_in[4],  (const float*)d_in[8],  (const float*)d_in[12]};
    const float* b1[NLAYERS] = {(const float*)d_in[5],  (const float*)d_in[9],  (const float*)d_in[13]};
    const float* w2[NLAYERS] = {(const float*)d_in[6],  (const float*)d_in[10], (const float*)d_in[14]};
    const float* b2[NLAYERS] = {(const float*)d_in[7],  (const float*)d_in[11], (const float*)d_in[15]};
    const float* w_out = (const float*)d_in[16];
    const float* b_out = (const float*)d_in[17];

    float* ws = (float*)d_ws;
    const size_t NH = (size_t)N_NODES * HD;
    float* agg = ws;                 // 25.6 MB
    float* t1  = ws + NH;            // hidden after MLP half 1
    float* hA  = ws + 2 * NH;        // layer output ping
    float* hB  = ws + 3 * NH;        // layer output pong
    float* ro  = ws + 4 * NH;        // [L][G][H] readouts (96 KB)

    hipMemsetAsync(ro, 0, (size_t)NLAYERS * NGRAPH * HD * sizeof(float), stream);

    const int mlp_grid  = N_NODES / 16;                 // 3125, exact
    const int scat_grid = (N_EDGES * 32 + 255) / 256;   // 100000 blocks
    const int ro_grid   = N_NODES / NODES_PER_BLOCK;    // 125, exact

    const float* hcur = h0;
    float* hnext = hA;
    for (int l = 0; l < NLAYERS; ++l) {
        hipMemsetAsync(agg, 0, NH * sizeof(float), stream);
        gin_edge_scatter<<<scat_grid, 256, 0, stream>>>(hcur, src, dst, agg);
        gin_mlp_wmma<<<mlp_grid, 256, 0, stream>>>(hcur, agg, w1[l], b1[l], t1, 1);
        gin_mlp_wmma<<<mlp_grid, 256, 0, stream>>>(t1, nullptr, w2[l], b2[l], hnext, 0);
        gin_readout<<<ro_grid, 128, 0, stream>>>(hnext, gids, ro + (size_t)l * NGRAPH * HD);
        hcur = hnext;
        hnext = (hnext == hA) ? hB : hA;
    }
    gin_final<<<1, 64, 0, stream>>>(ro, w_out, b_out, (float*)d_out);
}